// UniPre3D_Backbone_61538291417261
// MI455X (gfx1250) — compile-verified
//
#include <hip/hip_runtime.h>
#include <math.h>

typedef float v2f __attribute__((ext_vector_type(2)));
typedef float v8f __attribute__((ext_vector_type(8)));

static __device__ __forceinline__ v8f zero8() {
    v8f z = {0.f,0.f,0.f,0.f,0.f,0.f,0.f,0.f};
    return z;
}

// ---------------------------------------------------------------------------
// FPS: one block per batch, LDS distance array, block-wide argmax (first-index
// tie-break to match jnp.argmax). Sequential over M iterations.
// ---------------------------------------------------------------------------
__global__ void fps_kernel(const float* __restrict__ xyz, int Ncur, int M,
                           float* __restrict__ new_xyz, int* __restrict__ cent)
{
    __shared__ float s_dist[4096];
    __shared__ float s_rv[256];
    __shared__ int   s_ri[256];
    __shared__ int   s_far;
    int b = blockIdx.x, tid = threadIdx.x;
    const float* X = xyz + (size_t)b * Ncur * 3;
    for (int i = tid; i < Ncur; i += 256) s_dist[i] = 1e10f;
    if (tid == 0) s_far = 0;
    __syncthreads();
    for (int it = 0; it < M; ++it) {
        int far = s_far;
        float cx = X[far*3+0], cy = X[far*3+1], cz = X[far*3+2];
        if (tid == 0) {
            cent[b*M + it] = far;
            new_xyz[((size_t)b*M + it)*3 + 0] = cx;
            new_xyz[((size_t)b*M + it)*3 + 1] = cy;
            new_xyz[((size_t)b*M + it)*3 + 2] = cz;
        }
        float best = -1.f; int bi = 0;
        for (int i = tid; i < Ncur; i += 256) {
            float dx = X[i*3+0]-cx, dy = X[i*3+1]-cy, dz = X[i*3+2]-cz;
            float nd = fminf(s_dist[i], dx*dx + dy*dy + dz*dz);
            s_dist[i] = nd;
            if (nd > best) { best = nd; bi = i; }
        }
        s_rv[tid] = best; s_ri[tid] = bi;
        __syncthreads();
        for (int off = 128; off > 0; off >>= 1) {
            if (tid < off) {
                float ov = s_rv[tid+off]; int oi = s_ri[tid+off];
                if (ov > s_rv[tid] || (ov == s_rv[tid] && oi < s_ri[tid])) {
                    s_rv[tid] = ov; s_ri[tid] = oi;
                }
            }
            __syncthreads();
        }
        if (tid == 0) s_far = s_ri[0];
        __syncthreads();
    }
}

// ---------------------------------------------------------------------------
// KNN: one thread per query; 16-slot insertion sort; same d2 formula as ref.
// ---------------------------------------------------------------------------
__global__ void knn_kernel(const float* __restrict__ xyz, const float* __restrict__ qxyz,
                           int Ncur, int M, int Bn, int* __restrict__ outIdx)
{
    int t = blockIdx.x * blockDim.x + threadIdx.x;
    if (t >= Bn * M) return;
    int b = t / M;
    const float* X = xyz + (size_t)b * Ncur * 3;
    float qx = qxyz[(size_t)t*3+0], qy = qxyz[(size_t)t*3+1], qz = qxyz[(size_t)t*3+2];
    float qq = qx*qx + qy*qy + qz*qz;
    float bd[16]; int bix[16];
    #pragma unroll
    for (int j = 0; j < 16; ++j) { bd[j] = 3.4e38f; bix[j] = 0; }
    for (int n = 0; n < Ncur; ++n) {
        float px = X[n*3+0], py = X[n*3+1], pz = X[n*3+2];
        float pp = px*px + py*py + pz*pz;
        float d2 = qq + pp - 2.f*(qx*px + qy*py + qz*pz);
        if (d2 < bd[15]) {
            int j = 15;
            while (j > 0 && d2 < bd[j-1]) { bd[j] = bd[j-1]; bix[j] = bix[j-1]; --j; }
            bd[j] = d2; bix[j] = n;
        }
    }
    #pragma unroll
    for (int j = 0; j < 16; ++j) outIdx[(size_t)t*16 + j] = bix[j];
}

// ---------------------------------------------------------------------------
// Gather grouped xyz + features, concat into pin [B,M,16,3+prevD]
// ---------------------------------------------------------------------------
__global__ void gather_kernel(const float* __restrict__ cxyz, const float* __restrict__ cfeat,
                              const int* __restrict__ idx, float* __restrict__ pin,
                              int M, int Ncur, int prevD, size_t total)
{
    size_t t = (size_t)blockIdx.x * blockDim.x + threadIdx.x;
    if (t >= total) return;
    int in_dim = prevD + 3;
    int c = (int)(t % in_dim); size_t r = t / in_dim;
    int kk = (int)(r % 16); r /= 16;
    int m = (int)(r % M); int b = (int)(r / M);
    int nid = idx[((size_t)b*M + m)*16 + kk];
    float v;
    if (c < 3) v = cxyz[((size_t)b*Ncur + nid)*3 + c];
    else       v = cfeat[((size_t)b*Ncur + nid)*prevD + (c - 3)];
    pin[t] = v;
}

// ---------------------------------------------------------------------------
// WMMA f32 16x16x4 GEMM, 16x32 tile per wave, 4 waves per block.
// C = act(A[rows,K] @ B[K,N] + bias) + resid.
// Main K-loop is branch-free; ragged K handled in a single clamped-address
// remainder step (v_cndmask, no exec-mask branches).
// ---------------------------------------------------------------------------
__global__ void gemm_kernel(const float* __restrict__ A, const float* __restrict__ Bw,
                            const float* __restrict__ bias, const float* __restrict__ resid,
                            float* __restrict__ C, int K, int N, int relu)
{
    int lane = threadIdx.x & 31;
    int wid  = threadIdx.x >> 5;
    bool hi = lane >= 16;
    int l = lane & 15;
    int off2 = hi ? 2 : 0;
    int colT = blockIdx.x * 32;
    int rowT = (blockIdx.y * 4 + wid) * 16;
    const float* Arow = A + (size_t)(rowT + l) * K;
    const float* Bcol = Bw + colT + l;
    v8f acc0 = zero8(), acc1 = zero8();
    int kFull = K >> 2;
    for (int k4 = 0; k4 < kFull; ++k4) {
        int k0 = 4*k4 + off2;
        const float* bp = Bcol + (size_t)k0 * N;
        v2f a, b0, b1;
        a.x  = Arow[k0];  a.y  = Arow[k0+1];
        b0.x = bp[0];     b0.y = bp[N];
        b1.x = bp[16];    b1.y = bp[N+16];
        acc0 = __builtin_amdgcn_wmma_f32_16x16x4_f32(false, a, false, b0,
                                                     (short)0, acc0, false, false);
        acc1 = __builtin_amdgcn_wmma_f32_16x16x4_f32(false, a, false, b1,
                                                     (short)0, acc1, false, false);
    }
    if (K & 3) {
        int k0 = (K & ~3) + off2;
        int kc0 = (k0   < K) ? k0   : K-1;   // clamped addresses, unconditional loads
        int kc1 = (k0+1 < K) ? k0+1 : K-1;
        float a0 = Arow[kc0], a1 = Arow[kc1];
        const float* bp0 = Bcol + (size_t)kc0 * N;
        const float* bp1 = Bcol + (size_t)kc1 * N;
        float b00 = bp0[0], b01 = bp1[0], b10 = bp0[16], b11 = bp1[16];
        bool v0 = k0 < K, v1 = (k0+1) < K;
        v2f a, b0, b1;
        a.x  = v0 ? a0  : 0.f;  a.y  = v1 ? a1  : 0.f;
        b0.x = v0 ? b00 : 0.f;  b0.y = v1 ? b01 : 0.f;
        b1.x = v0 ? b10 : 0.f;  b1.y = v1 ? b11 : 0.f;
        acc0 = __builtin_amdgcn_wmma_f32_16x16x4_f32(false, a, false, b0,
                                                     (short)0, acc0, false, false);
        acc1 = __builtin_amdgcn_wmma_f32_16x16x4_f32(false, a, false, b1,
                                                     (short)0, acc1, false, false);
    }
    int col0 = colT + l, col1 = colT + 16 + l;
    float bb0 = bias ? bias[col0] : 0.f;
    float bb1 = bias ? bias[col1] : 0.f;
    #pragma unroll
    for (int r = 0; r < 8; ++r) {
        int row = rowT + r + (hi ? 8 : 0);
        float v0 = acc0[r] + bb0;
        float v1 = acc1[r] + bb1;
        if (relu) { v0 = fmaxf(v0, 0.f); v1 = fmaxf(v1, 0.f); }
        if (resid) {
            v0 += resid[(size_t)row*N + col0];
            v1 += resid[(size_t)row*N + col1];
        }
        C[(size_t)row*N + col0] = v0;
        C[(size_t)row*N + col1] = v1;
    }
}

// ---------------------------------------------------------------------------
// Same GEMM, epilogue reduces max over the 16 rows of the tile
// (= max over the K_NN=16 neighbors of one point). Output row per wave.
// ---------------------------------------------------------------------------
__global__ void gemm_rowmax_kernel(const float* __restrict__ A, const float* __restrict__ Bw,
                                   const float* __restrict__ bias, float* __restrict__ Cout,
                                   int K, int N)
{
    int lane = threadIdx.x & 31;
    int wid  = threadIdx.x >> 5;
    bool hi = lane >= 16;
    int l = lane & 15;
    int off2 = hi ? 2 : 0;
    int colT = blockIdx.x * 32;
    int outRow = blockIdx.y * 4 + wid;
    int rowT = outRow * 16;
    const float* Arow = A + (size_t)(rowT + l) * K;
    const float* Bcol = Bw + colT + l;
    v8f acc0 = zero8(), acc1 = zero8();
    int kFull = K >> 2;   // K here is d: always a multiple of 4
    for (int k4 = 0; k4 < kFull; ++k4) {
        int k0 = 4*k4 + off2;
        const float* bp = Bcol + (size_t)k0 * N;
        v2f a, b0, b1;
        a.x  = Arow[k0];  a.y  = Arow[k0+1];
        b0.x = bp[0];     b0.y = bp[N];
        b1.x = bp[16];    b1.y = bp[N+16];
        acc0 = __builtin_amdgcn_wmma_f32_16x16x4_f32(false, a, false, b0,
                                                     (short)0, acc0, false, false);
        acc1 = __builtin_amdgcn_wmma_f32_16x16x4_f32(false, a, false, b1,
                                                     (short)0, acc1, false, false);
    }
    int col0 = colT + l, col1 = colT + 16 + l;
    float bb0 = bias ? bias[col0] : 0.f;
    float bb1 = bias ? bias[col1] : 0.f;
    float mx0 = -3.4e38f, mx1 = -3.4e38f;
    #pragma unroll
    for (int r = 0; r < 8; ++r) {
        mx0 = fmaxf(mx0, acc0[r] + bb0);
        mx1 = fmaxf(mx1, acc1[r] + bb1);
    }
    mx0 = fmaxf(mx0, __shfl_xor(mx0, 16, 32));   // combine the two row-halves
    mx1 = fmaxf(mx1, __shfl_xor(mx1, 16, 32));
    if (!hi) {
        Cout[(size_t)outRow * N + col0] = mx0;
        Cout[(size_t)outRow * N + col1] = mx1;
    }
}

// ---------------------------------------------------------------------------
// LayerNorm: one wave per row (4 rows per 128-thread block)
// ---------------------------------------------------------------------------
__global__ void layernorm_kernel(const float* __restrict__ X, const float* __restrict__ g,
                                 const float* __restrict__ be, float* __restrict__ Y,
                                 int rows, int D)
{
    int lane = threadIdx.x & 31, wv = threadIdx.x >> 5;
    int row = blockIdx.x * 4 + wv;
    if (row >= rows) return;
    const float* x = X + (size_t)row * D;
    float s = 0.f, s2 = 0.f;
    for (int c = lane; c < D; c += 32) { float v = x[c]; s += v; s2 += v*v; }
    for (int m = 1; m < 32; m <<= 1) { s += __shfl_xor(s, m, 32); s2 += __shfl_xor(s2, m, 32); }
    float mean = s / D;
    float var = fmaxf(s2 / D - mean*mean, 0.f);
    float rs = rsqrtf(var + 1e-5f);
    float* y = Y + (size_t)row * D;
    for (int c = lane; c < D; c += 32) y[c] = (x[c] - mean) * rs * g[c] + be[c];
}

// ---------------------------------------------------------------------------
// Fused flash attention, WMMA f32 16x16x4 for both Q.K^T and P.V.
// One wave per (16-query tile, b, h). Q/K/V/O layout: [B, S, H, DH].
// ---------------------------------------------------------------------------
template<int DH>
__global__ void attn_kernel(const float* __restrict__ Q, const float* __restrict__ Kx,
                            const float* __restrict__ V, float* __restrict__ O, int S)
{
    constexpr int H = 4;
    constexpr int DH4 = DH / 4;
    constexpr int DHT = DH / 16;
    __shared__ float P[16][17];
    int lane = threadIdx.x;
    bool hi = lane >= 16;
    int l = lane & 15;
    int q0 = blockIdx.x * 16;
    int bh = blockIdx.y;
    int b = bh / H, h = bh % H;
    const float scale = 1.0f / sqrtf((float)DH);
    const size_t sStride = (size_t)H * DH;
    size_t base = (size_t)b * S * sStride + (size_t)h * DH;

    v2f a[DH4];
    #pragma unroll
    for (int k4 = 0; k4 < DH4; ++k4) {
        int k0 = 4*k4 + (hi ? 2 : 0);
        const float* q = Q + base + (size_t)(q0 + l) * sStride;
        a[k4].x = q[k0]; a[k4].y = q[k0+1];
    }
    v8f o[DHT];
    #pragma unroll
    for (int n = 0; n < DHT; ++n) o[n] = zero8();
    float mrow[8], lrow[8];
    #pragma unroll
    for (int r = 0; r < 8; ++r) { mrow[r] = -3.4e38f; lrow[r] = 0.f; }

    int nKT = S / 16;
    for (int kt = 0; kt < nKT; ++kt) {
        int s0 = kt * 16;
        v8f sc = zero8();
        #pragma unroll
        for (int k4 = 0; k4 < DH4; ++k4) {
            int k0 = 4*k4 + (hi ? 2 : 0);
            const float* kp = Kx + base + (size_t)(s0 + l) * sStride;
            v2f bf; bf.x = kp[k0]; bf.y = kp[k0+1];
            sc = __builtin_amdgcn_wmma_f32_16x16x4_f32(false, a[k4], false, bf,
                                                       (short)0, sc, false, false);
        }
        #pragma unroll
        for (int r = 0; r < 8; ++r) {
            float s = sc[r] * scale;
            float rm = s;
            rm = fmaxf(rm, __shfl_xor(rm, 1, 32));
            rm = fmaxf(rm, __shfl_xor(rm, 2, 32));
            rm = fmaxf(rm, __shfl_xor(rm, 4, 32));
            rm = fmaxf(rm, __shfl_xor(rm, 8, 32));
            float mn = fmaxf(mrow[r], rm);
            float alpha = expf(mrow[r] - mn);
            float p = expf(s - mn);
            float rs = p;
            rs += __shfl_xor(rs, 1, 32);
            rs += __shfl_xor(rs, 2, 32);
            rs += __shfl_xor(rs, 4, 32);
            rs += __shfl_xor(rs, 8, 32);
            lrow[r] = lrow[r] * alpha + rs;
            mrow[r] = mn;
            #pragma unroll
            for (int n = 0; n < DHT; ++n) o[n][r] *= alpha;
            P[r + (hi ? 8 : 0)][l] = p;
        }
        __syncthreads();
        v2f pa[4];
        #pragma unroll
        for (int k4 = 0; k4 < 4; ++k4) {
            int k0 = 4*k4 + (hi ? 2 : 0);
            pa[k4].x = P[l][k0]; pa[k4].y = P[l][k0+1];
        }
        #pragma unroll
        for (int n = 0; n < DHT; ++n) {
            #pragma unroll
            for (int k4 = 0; k4 < 4; ++k4) {
                int kk = s0 + 4*k4 + (hi ? 2 : 0);
                const float* vp = V + base + (size_t)kk * sStride + n*16 + l;
                v2f bf; bf.x = vp[0]; bf.y = vp[sStride];
                o[n] = __builtin_amdgcn_wmma_f32_16x16x4_f32(false, pa[k4], false, bf,
                                                             (short)0, o[n], false, false);
            }
        }
        __syncthreads();
    }
    #pragma unroll
    for (int n = 0; n < DHT; ++n) {
        #pragma unroll
        for (int r = 0; r < 8; ++r) {
            int row = q0 + r + (hi ? 8 : 0);
            O[base + (size_t)row * sStride + n*16 + l] = o[n][r] / lrow[r];
        }
    }
}

// ---------------------------------------------------------------------------
// Mean over rows of feats[3]
// ---------------------------------------------------------------------------
__global__ void mean_kernel(const float* __restrict__ X, float* __restrict__ out,
                            int Mrows, int D, int Bn)
{
    int t = blockIdx.x * blockDim.x + threadIdx.x;
    if (t >= Bn * D) return;
    int b = t / D, c = t % D;
    float s = 0.f;
    for (int m = 0; m < Mrows; ++m) s += X[((size_t)b*Mrows + m)*D + c];
    out[t] = s / (float)Mrows;
}

// ---------------------------------------------------------------------------
extern "C" void kernel_launch(void* const* d_in, const int* in_sizes, int n_in,
                              void* d_out, int out_size, void* d_ws, size_t ws_size,
                              hipStream_t stream)
{
    (void)in_sizes; (void)n_in; (void)out_size; (void)ws_size;
    const int B = 2, N0 = 4096, KNN = 16;
    const int DIMS[4] = {64, 128, 256, 512};
    const float* xyz = (const float*)d_in[0];

    // params flattened jax-style: list order, dict keys sorted
    enum { pWk=0,pWo,pWq,pWv,pbk,pbo,pbq,pbv,pln1_b,pln1_g,pln2_b,pln2_g,
           pmW1,pmW2,pmb1,pmb2,ppW1,ppW2,ppb1,ppb2 };
    auto P = [&](int st, int id) -> const float* {
        return (const float*)d_in[1 + st*20 + id];
    };

    float* W = (float*)d_ws;
    size_t off = 0;
    auto alloc = [&](size_t n) -> size_t {
        size_t o = off; off += (n + 63) & ~(size_t)63; return o;
    };
    size_t o_xyzA = alloc((size_t)B*N0*3);
    size_t o_xyzB = alloc((size_t)B*N0*3);
    size_t o_cent = alloc((size_t)B*N0);        // int
    size_t o_knn  = alloc((size_t)B*N0*KNN);    // int
    size_t o_pin  = alloc(2195456);             // max B*M*16*(prevD+3) (stage 1)
    size_t o_h1   = alloc(8388608);             // max B*M*16*d (stage 0)
    size_t o_emb  = alloc(524288);
    size_t o_ln1  = alloc(524288);
    size_t o_q    = alloc(524288);
    size_t o_k    = alloc(524288);
    size_t o_v    = alloc(524288);
    size_t o_ao   = alloc(524288);
    size_t o_x    = alloc(524288);
    size_t o_ln2  = alloc(524288);
    size_t o_ffn  = alloc(2097152);             // max B*M*4d (stage 0)

    float* out = (float*)d_out;
    size_t fOff[4] = {0, 524288, 786432, 917504};
    size_t meanOff = 983040;

    const float* cur_xyz = xyz;
    const float* cur_feat = xyz;
    int Ncur = N0, prevD = 3;

    for (int st = 0; st < 4; ++st) {
        int M = N0 >> (2*st);
        int d = DIMS[st];
        int in_dim = prevD + 3;
        int hid = 4 * d;
        float* new_xyz = W + ((st & 1) ? o_xyzB : o_xyzA);

        fps_kernel<<<B, 256, 0, stream>>>(cur_xyz, Ncur, M, new_xyz, (int*)(W + o_cent));

        int totQ = B * M;
        knn_kernel<<<(totQ + 127)/128, 128, 0, stream>>>(cur_xyz, new_xyz, Ncur, M, B,
                                                         (int*)(W + o_knn));

        size_t totG = (size_t)B * M * KNN * in_dim;
        gather_kernel<<<(unsigned)((totG + 255)/256), 256, 0, stream>>>(
            cur_xyz, cur_feat, (int*)(W + o_knn), W + o_pin, M, Ncur, prevD, totG);

        int rowsP = B * M * KNN;
        gemm_kernel<<<dim3(d/32, rowsP/64), 128, 0, stream>>>(
            W + o_pin, P(st,ppW1), P(st,ppb1), nullptr, W + o_h1, in_dim, d, 1);
        gemm_rowmax_kernel<<<dim3(d/32, rowsP/64), 128, 0, stream>>>(
            W + o_h1, P(st,ppW2), P(st,ppb2), W + o_emb, d, d);

        int Srows = B * M;
        layernorm_kernel<<<(Srows + 3)/4, 128, 0, stream>>>(
            W + o_emb, P(st,pln1_g), P(st,pln1_b), W + o_ln1, Srows, d);

        gemm_kernel<<<dim3(d/32, Srows/64), 128, 0, stream>>>(
            W + o_ln1, P(st,pWq), P(st,pbq), nullptr, W + o_q, d, d, 0);
        gemm_kernel<<<dim3(d/32, Srows/64), 128, 0, stream>>>(
            W + o_ln1, P(st,pWk), P(st,pbk), nullptr, W + o_k, d, d, 0);
        gemm_kernel<<<dim3(d/32, Srows/64), 128, 0, stream>>>(
            W + o_ln1, P(st,pWv), P(st,pbv), nullptr, W + o_v, d, d, 0);

        int S = M;
        dim3 ag(S/16, B*4);
        switch (d / 4) {
            case 16:  attn_kernel<16> <<<ag, 32, 0, stream>>>(W+o_q, W+o_k, W+o_v, W+o_ao, S); break;
            case 32:  attn_kernel<32> <<<ag, 32, 0, stream>>>(W+o_q, W+o_k, W+o_v, W+o_ao, S); break;
            case 64:  attn_kernel<64> <<<ag, 32, 0, stream>>>(W+o_q, W+o_k, W+o_v, W+o_ao, S); break;
            case 128: attn_kernel<128><<<ag, 32, 0, stream>>>(W+o_q, W+o_k, W+o_v, W+o_ao, S); break;
        }

        // x = attn_o @ Wo + bo + embed
        gemm_kernel<<<dim3(d/32, Srows/64), 128, 0, stream>>>(
            W + o_ao, P(st,pWo), P(st,pbo), W + o_emb, W + o_x, d, d, 0);

        layernorm_kernel<<<(Srows + 3)/4, 128, 0, stream>>>(
            W + o_x, P(st,pln2_g), P(st,pln2_b), W + o_ln2, Srows, d);

        gemm_kernel<<<dim3(hid/32, Srows/64), 128, 0, stream>>>(
            W + o_ln2, P(st,pmW1), P(st,pmb1), nullptr, W + o_ffn, d, hid, 1);
        // feats[st] = ffn @ mW2 + mb2 + x  -> written straight into d_out
        gemm_kernel<<<dim3(d/32, Srows/64), 128, 0, stream>>>(
            W + o_ffn, P(st,pmW2), P(st,pmb2), W + o_x, out + fOff[st], hid, d, 0);

        cur_xyz = new_xyz;
        cur_feat = out + fOff[st];
        Ncur = M;
        prevD = d;
    }

    mean_kernel<<<(B*512 + 255)/256, 256, 0, stream>>>(out + fOff[3], out + meanOff, 64, 512, B);
}